// HashEncode2D_87986700026016
// MI455X (gfx1250) — compile-verified
//
#include <hip/hip_runtime.h>

// HashEncode2D for MI455X (gfx1250, wave32).
// Gather-latency-bound kernel: restructured so each wave keeps all 64
// table gathers in flight (single load phase), then consumes them in
// order (VMEM loads return in order -> decreasing s_wait_loadcnt).
// No matmul structure -> WMMA not applicable; optimize the memory path.

#define LEVELS 16
#define TBITS  19
#define TMASK  ((1u << TBITS) - 1u)
#define PRIME2 2654435761u
#define HW     512
#define PLANE  (HW * HW)        // 262144 points per image plane

__global__ __launch_bounds__(256)
void hash_encode2d_kernel(const float* __restrict__ coord,
                          const float* __restrict__ table,
                          float* __restrict__ out)
{
    // 2D tile: 32 (w) x 8 (h) threads per block; a wave32 = one 32-pixel row.
    const int lane = threadIdx.x & 31;
    const int row  = threadIdx.x >> 5;
    const int w = (blockIdx.x << 5) + lane;
    const int h = (blockIdx.y << 3) + row;
    const int b = blockIdx.z;
    const int p = h * HW + w;

    // coord layout: (4, 2, 512, 512); streaming read -> non-temporal.
    const float* cb = coord + (size_t)b * (2 * PLANE);
    const float cx = __builtin_nontemporal_load(cb + p);
    const float cy = __builtin_nontemporal_load(cb + PLANE + p);

    // out layout: (4, 32, 512, 512); plane stride PLANE floats.
    float* ob = out + (size_t)b * (2 * LEVELS) * PLANE + p;

    // NS = int(8 * (16^(1/15))^i), i = 0..15
    const int ns_arr[LEVELS] = {8, 9, 11, 13, 16, 20, 24, 29,
                                35, 42, 50, 61, 73, 88, 106, 128};

    // ---------------- Phase 1: issue ALL 64 gathers ----------------
    // f[l][k] lives in registers (fully unrolled); 128 VGPRs of payload.
    float2 f[LEVELS][4];

#pragma unroll
    for (int l = 0; l < LEVELS; ++l) {
        const float n = (float)ns_arr[l];
        const int hx = (int)(cx * n);
        const int hy = (int)(cy * n);

        const uint32_t hyp0 = (uint32_t)hy * PRIME2;
        const uint32_t hyp1 = (uint32_t)(hy + 1) * PRIME2;
        const uint32_t i1 = (((uint32_t)hx)       ^ hyp0) & TMASK;
        const uint32_t i2 = (((uint32_t)(hx + 1)) ^ hyp0) & TMASK;
        const uint32_t i3 = (((uint32_t)hx)       ^ hyp1) & TMASK;
        const uint32_t i4 = (((uint32_t)(hx + 1)) ^ hyp1) & TMASK;

        // table layout: (16, 2^19, 2) f32 -> float2 gathers (global_load_b64).
        const float2* tl = (const float2*)table + ((size_t)l << TBITS);
        f[l][0] = tl[i1];
        f[l][1] = tl[i2];
        f[l][2] = tl[i3];
        f[l][3] = tl[i4];
    }

    // ---------------- Phase 2: consume in order, store NT ----------------
#pragma unroll
    for (int l = 0; l < LEVELS; ++l) {
        const float n   = (float)ns_arr[l];
        const float inv = 1.0f / n;

        // cx >= 0, so floor(cx*n) == trunc(cx*n); ulp-equivalent to the
        // reference's floor(cx / inv) without a 10-op f32 division.
        const float bx = floorf(cx * n);
        const float by = floorf(cy * n);

        const float wx_lo = ((bx + 1.0f) * inv - cx) * n;
        const float wx_hi = (cx - bx * inv) * n;
        const float wy_lo = ((by + 1.0f) * inv - cy) * n;
        const float wy_hi = (cy - by * inv) * n;

        const float2 f1 = f[l][0];
        const float2 f2 = f[l][1];
        const float2 f3 = f[l][2];
        const float2 f4 = f[l][3];

        const float r1x = f1.x * wx_lo + f2.x * wx_hi;
        const float r1y = f1.y * wx_lo + f2.y * wx_hi;
        const float r2x = f3.x * wx_lo + f4.x * wx_hi;
        const float r2y = f3.y * wx_lo + f4.y * wx_hi;

        const float o0 = r1x * wy_lo + r2x * wy_hi;
        const float o1 = r1y * wy_lo + r2y * wy_hi;

        // Output is write-once streaming: NT stores keep L2 free for the
        // 64 MiB hash table (fits in the 192 MB global L2).
        __builtin_nontemporal_store(o0, ob + (size_t)(2 * l)     * PLANE);
        __builtin_nontemporal_store(o1, ob + (size_t)(2 * l + 1) * PLANE);
    }
}

extern "C" void kernel_launch(void* const* d_in, const int* in_sizes, int n_in,
                              void* d_out, int out_size, void* d_ws, size_t ws_size,
                              hipStream_t stream)
{
    const float* coord = (const float*)d_in[0];   // (4, 2, 512, 512) f32
    const float* table = (const float*)d_in[1];   // (16, 2^19, 2) f32
    float* out = (float*)d_out;                   // (4, 32, 512, 512) f32

    dim3 grid(HW / 32, HW / 8, 4);                // 16 x 64 x 4 blocks
    dim3 block(256);
    hash_encode2d_kernel<<<grid, block, 0, stream>>>(coord, table, out);
}